// RelativeMultiHeadAttention_21363167330655
// MI455X (gfx1250) — compile-verified
//
#include <hip/hip_runtime.h>

// ---------------------------------------------------------------------------
// Transformer-XL relative multi-head attention, fused for gfx1250 (MI455X).
// All GEMMs use V_WMMA_F32_16X16X4_F32 (full f32 numerics; problem is
// bandwidth-bound at 24 GFLOP / ~60MB so f32 matrix ops are the right choice).
// ---------------------------------------------------------------------------

#define DIMN   512
#define NHEAD  16
#define DH     32
#define BATCH  4
#define SEQ    1024
#define MTOT   (BATCH * SEQ)   // 4096 rows

typedef __attribute__((ext_vector_type(2))) float v2f;
typedef __attribute__((ext_vector_type(8))) float v8f;

// D = A(16x4) * B(4x16) + C, f32, wave32.
__device__ __forceinline__ v8f wmma_f32(v2f a, v2f b, v8f c) {
  return __builtin_amdgcn_wmma_f32_16x16x4_f32(
      /*neg_a=*/false, a, /*neg_b=*/false, b,
      /*c_mod=*/(short)0, c, /*reuse_a=*/false, /*reuse_b=*/false);
}

// ---------------------------------------------------------------------------
// One wave computes a 16(M) x 64(N) tile of  out = X @ W^T (+ bias).
// X: [M, K] row-major, W: [N, K] row-major (torch Linear weight), K = DIMN.
// B-matrix column n == W row n -> contiguous float2 loads along K.
// ---------------------------------------------------------------------------
__device__ __forceinline__ void gemm_tile_16x64(
    const float* __restrict__ X, const float* __restrict__ W,
    const float* __restrict__ bias, float* __restrict__ out,
    int m0, int n0, int lane) {
  const int half = lane >> 4;   // K sub-slot (ISA: VGPR0 holds K=2*half)
  const int lm   = lane & 15;   // M row (A) / N col (B)

  const float* arow = X + (size_t)(m0 + lm) * DIMN;
  const float* brow = W + (size_t)(n0 + lm) * DIMN;

  v8f acc[4] = {v8f{}, v8f{}, v8f{}, v8f{}};

  for (int kk = 0; kk < DIMN; kk += 4) {
    const int ko = kk + 2 * half;
    v2f a  = *(const v2f*)(arow + ko);
    v2f b0 = *(const v2f*)(brow + 0 * 16 * DIMN + ko);
    v2f b1 = *(const v2f*)(brow + 1 * 16 * DIMN + ko);
    v2f b2 = *(const v2f*)(brow + 2 * 16 * DIMN + ko);
    v2f b3 = *(const v2f*)(brow + 3 * 16 * DIMN + ko);
    acc[0] = wmma_f32(a, b0, acc[0]);
    acc[1] = wmma_f32(a, b1, acc[1]);
    acc[2] = wmma_f32(a, b2, acc[2]);
    acc[3] = wmma_f32(a, b3, acc[3]);
  }

  #pragma unroll
  for (int t = 0; t < 4; ++t) {
    const int n = n0 + 16 * t + lm;
    const float bval = bias ? bias[n] : 0.0f;
    #pragma unroll
    for (int j = 0; j < 8; ++j) {           // D layout: M = j + 8*half, N = lm
      const int m = m0 + j + 8 * half;
      out[(size_t)m * DIMN + n] = acc[t][j] + bval;
    }
  }
}

// ---------------------------------------------------------------------------
// Kernel A: all four input projections (q,k,v,pos) in one launch.
// 8192 waves: [matrix(4)][mtile(256)][ntile(8)].
// ---------------------------------------------------------------------------
__global__ __launch_bounds__(256) void proj_kernel(
    const float* __restrict__ q,  const float* __restrict__ k,
    const float* __restrict__ v,  const float* __restrict__ p,
    const float* __restrict__ Wq, const float* __restrict__ bq,
    const float* __restrict__ Wk, const float* __restrict__ bk,
    const float* __restrict__ Wv, const float* __restrict__ bvp,
    const float* __restrict__ Wp,
    float* __restrict__ qp, float* __restrict__ kp,
    float* __restrict__ vp, float* __restrict__ pp) {
  const int wid  = (blockIdx.x * blockDim.x + threadIdx.x) >> 5;
  const int lane = threadIdx.x & 31;
  const int mat  = wid >> 11;          // 0..3
  const int rem  = wid & 2047;
  const int mt   = rem >> 3;           // 0..255
  const int nt   = rem & 7;            // 0..7

  const float *X, *W, *bias; float* out;
  switch (mat) {
    case 0:  X = q; W = Wq; bias = bq;     out = qp; break;
    case 1:  X = k; W = Wk; bias = bk;     out = kp; break;
    case 2:  X = v; W = Wv; bias = bvp;    out = vp; break;
    default: X = p; W = Wp; bias = nullptr; out = pp; break;
  }
  gemm_tile_16x64(X, W, bias, out, mt * 16, nt * 64, lane);
}

// ---------------------------------------------------------------------------
// Kernel B: fused relative attention. One workgroup (4 waves) per
// (batch, head, 16-row q-tile). grid = 4*16*64 = 4096 blocks.
//
// Relative shift (derived from the pad/reshape reference):
//   shifted[q,k] = PS[q][S-1+k-q]      if k <= q
//                = 0                   if k == q+1
//                = PS[q+1][k-q-2]      if k >= q+2
// so we need PS rows q0..q0+16 (17 rows) over all 1024 columns -> LDS.
// ---------------------------------------------------------------------------
__global__ __launch_bounds__(128) void attn_kernel(
    const float* __restrict__ qp, const float* __restrict__ kp,
    const float* __restrict__ vp, const float* __restrict__ pp,
    const float* __restrict__ u_bias, const float* __restrict__ v_bias,
    float* __restrict__ ctx) {
  const int qt  = blockIdx.x & 63;
  const int h   = (blockIdx.x >> 6) & 15;
  const int bat = blockIdx.x >> 10;
  const int q0  = qt * 16;

  const int tid  = threadIdx.x;
  const int wave = tid >> 5;
  const int lane = tid & 31;
  const int half = lane >> 4;
  const int lm   = lane & 15;

  __shared__ float PS[17][SEQ];        // pos scores, rows q0..q0+16   (68 KB)
  __shared__ float SC[16][SEQ];        // scores -> exp(scores)        (64 KB)
  __shared__ float inv_sum[16];
  __shared__ float ctxp[4][16][DH];    // per-wave partial context      (8 KB)

  const size_t base = (size_t)bat * SEQ;
  const int hd = h * DH;

  // Hoisted A-fragments for this wave's q rows: (q + u_bias), (q + v_bias).
  v2f aqu[8], aqv[8];
  {
    const float* qrow = qp + (base + q0 + lm) * DIMN + hd;
    #pragma unroll
    for (int j = 0; j < 8; ++j) {
      const int d = 4 * j + 2 * half;
      const float x0 = qrow[d], x1 = qrow[d + 1];
      v2f u; u.x = x0 + u_bias[hd + d]; u.y = x1 + u_bias[hd + d + 1];
      v2f w; w.x = x0 + v_bias[hd + d]; w.y = x1 + v_bias[hd + d + 1];
      aqu[j] = u; aqv[j] = w;
    }
  }

  // ---- Phase 1: PS rows 0..15 via WMMA; row 16 via VALU dot products. ----
  for (int ct = wave; ct < SEQ / 16; ct += 4) {
    const int c0 = ct * 16;
    const float* prow = pp + (base + c0 + lm) * DIMN + hd;
    v8f acc = {};
    #pragma unroll
    for (int j = 0; j < 8; ++j) {
      v2f bfr = *(const v2f*)(prow + 4 * j + 2 * half);
      acc = wmma_f32(aqv[j], bfr, acc);
    }
    #pragma unroll
    for (int j = 0; j < 8; ++j) PS[j + 8 * half][c0 + lm] = acc[j];
  }
  if (q0 + 16 < SEQ) {
    const float* qrow = qp + (base + q0 + 16) * DIMN + hd;
    for (int c = tid; c < SEQ; c += 128) {
      const float* prow = pp + (base + c) * DIMN + hd;
      float s = 0.f;
      #pragma unroll
      for (int d = 0; d < DH; ++d) s += (qrow[d] + v_bias[hd + d]) * prow[d];
      PS[16][c] = s;
    }
  } else {
    for (int c = tid; c < SEQ; c += 128) PS[16][c] = 0.f;
  }
  __syncthreads();

  // ---- Phase 2: content scores via WMMA + shifted pos + scale. ----
  const float scale = 0.17677669529663687f;   // 1/sqrt(32)
  for (int kt = wave; kt < SEQ / 16; kt += 4) {
    const int k0 = kt * 16;
    const float* krow = kp + (base + k0 + lm) * DIMN + hd;
    v8f acc = {};
    #pragma unroll
    for (int j = 0; j < 8; ++j) {
      v2f bfr = *(const v2f*)(krow + 4 * j + 2 * half);
      acc = wmma_f32(aqu[j], bfr, acc);
    }
    const int gk = k0 + lm;
    #pragma unroll
    for (int j = 0; j < 8; ++j) {
      const int rq = j + 8 * half;
      const int gq = q0 + rq;
      float pos;
      if (gk <= gq)            pos = PS[rq][SEQ - 1 + gk - gq];
      else if (gk == gq + 1)   pos = 0.f;
      else                     pos = PS[rq + 1][gk - gq - 2];
      SC[rq][gk] = (acc[j] + pos) * scale;
    }
  }
  __syncthreads();

  // ---- Phase 3: softmax per row (wave handles 4 rows, lane-parallel). ----
  for (int r = wave * 4; r < wave * 4 + 4; ++r) {
    float m = -3.402823466e38f;
    float xv[32];
    #pragma unroll
    for (int i = 0; i < 32; ++i) { xv[i] = SC[r][lane + 32 * i]; m = fmaxf(m, xv[i]); }
    #pragma unroll
    for (int o = 16; o > 0; o >>= 1) m = fmaxf(m, __shfl_xor(m, o, 32));
    float s = 0.f;
    #pragma unroll
    for (int i = 0; i < 32; ++i) {
      const float e = __expf(xv[i] - m);
      SC[r][lane + 32 * i] = e;
      s += e;
    }
    #pragma unroll
    for (int o = 16; o > 0; o >>= 1) s += __shfl_xor(s, o, 32);
    if (lane == 0) inv_sum[r] = 1.0f / s;
  }
  __syncthreads();

  // ---- Phase 4: context = P @ V via WMMA (A re-read from LDS in A-layout). --
  v8f cd0 = {}, cd1 = {};
  for (int kt = wave * 16; kt < wave * 16 + 16; ++kt) {
    const int k0 = kt * 16;
    #pragma unroll
    for (int j = 0; j < 4; ++j) {
      const int kb = k0 + 4 * j + 2 * half;
      v2f a = *(const v2f*)(&SC[lm][kb]);
      const float* v0 = vp + (base + kb) * DIMN + hd;
      const float* v1 = vp + (base + kb + 1) * DIMN + hd;
      v2f b0; b0.x = v0[lm];      b0.y = v1[lm];
      v2f b1; b1.x = v0[16 + lm]; b1.y = v1[16 + lm];
      cd0 = wmma_f32(a, b0, cd0);
      cd1 = wmma_f32(a, b1, cd1);
    }
  }
  #pragma unroll
  for (int j = 0; j < 8; ++j) {
    ctxp[wave][j + 8 * half][lm]      = cd0[j];
    ctxp[wave][j + 8 * half][16 + lm] = cd1[j];
  }
  __syncthreads();
  for (int e = tid; e < 16 * DH; e += 128) {
    const int r = e >> 5, d = e & 31;
    const float s = ctxp[0][r][d] + ctxp[1][r][d] + ctxp[2][r][d] + ctxp[3][r][d];
    ctx[(base + q0 + r) * DIMN + hd + d] = s * inv_sum[r];
  }
}

// ---------------------------------------------------------------------------
// Kernel C: out = ctx @ Wo^T + bo. 2048 waves.
// ---------------------------------------------------------------------------
__global__ __launch_bounds__(256) void out_kernel(
    const float* __restrict__ ctx, const float* __restrict__ Wo,
    const float* __restrict__ bo, float* __restrict__ out) {
  const int wid  = (blockIdx.x * blockDim.x + threadIdx.x) >> 5;
  const int lane = threadIdx.x & 31;
  const int mt = wid >> 3, nt = wid & 7;
  gemm_tile_16x64(ctx, Wo, bo, out, mt * 16, nt * 64, lane);
}

// ---------------------------------------------------------------------------
extern "C" void kernel_launch(void* const* d_in, const int* in_sizes, int n_in,
                              void* d_out, int out_size, void* d_ws, size_t ws_size,
                              hipStream_t stream) {
  (void)in_sizes; (void)n_in; (void)out_size; (void)ws_size;

  const float* query = (const float*)d_in[0];
  const float* key   = (const float*)d_in[1];
  const float* value = (const float*)d_in[2];
  const float* pos   = (const float*)d_in[3];
  const float* Wq    = (const float*)d_in[4];
  const float* bq    = (const float*)d_in[5];
  const float* Wk    = (const float*)d_in[6];
  const float* bk    = (const float*)d_in[7];
  const float* Wv    = (const float*)d_in[8];
  const float* bvb   = (const float*)d_in[9];
  const float* Wp    = (const float*)d_in[10];
  const float* u_b   = (const float*)d_in[11];
  const float* v_b   = (const float*)d_in[12];
  const float* Wo    = (const float*)d_in[13];
  const float* bo    = (const float*)d_in[14];

  const size_t MAT = (size_t)MTOT * DIMN;   // 2,097,152 floats (8 MB)
  float* qp  = (float*)d_ws;
  float* kp  = qp + MAT;
  float* vp  = kp + MAT;
  float* pp  = vp + MAT;
  float* ctx = pp + MAT;                    // 40 MB total scratch

  proj_kernel<<<1024, 256, 0, stream>>>(query, key, value, pos,
                                        Wq, bq, Wk, bk, Wv, bvb, Wp,
                                        qp, kp, vp, pp);
  attn_kernel<<<BATCH * NHEAD * (SEQ / 16), 128, 0, stream>>>(
      qp, kp, vp, pp, u_b, v_b, ctx);
  out_kernel<<<256, 256, 0, stream>>>(ctx, Wo, bo, (float*)d_out);
}